// MaSIF_site_38474317037712
// MI455X (gfx1250) — compile-verified
//
#include <hip/hip_runtime.h>
#include <hip/hip_bf16.h>
#include <math.h>

typedef float v2f __attribute__((ext_vector_type(2)));
typedef float v8f __attribute__((ext_vector_type(8)));

#define NSAMP   10000
#define NVERT   128
#define NFEAT   5
#define NGAUSS  80
#define NROT    16
#define TILE_N  16
#define KCHUNK  2
#define DIN     (NFEAT * NGAUSS)   // 400
#define TWO_PI_F 6.283185307179586f
#define DTHETA   (TWO_PI_F / 16.0f)
#define EPSF     1e-5f

// -------------------------------------------------------------------------
// Kernel 1: per-feature gaussian soft-binning + 80x80 conv + rotation-max
// grid = (625 sample tiles, 5 features), block = 256 threads (8 wave32)
// -------------------------------------------------------------------------
__global__ __launch_bounds__(256) void masif_gauss_conv_kernel(
    const float* __restrict__ rho,      const float* __restrict__ theta,
    const float* __restrict__ feat,     const float* __restrict__ mask,
    const float* __restrict__ mu_rho,   const float* __restrict__ sigma_rho,
    const float* __restrict__ mu_theta, const float* __restrict__ sigma_theta,
    const float* __restrict__ Wc,       const float* __restrict__ bc,
    float* __restrict__ x_out)
{
    __shared__ float s_rho [TILE_N * NVERT];      // 8 KB (mask folded in)
    __shared__ float s_th  [TILE_N * NVERT];      // 8 KB
    __shared__ float s_ft  [TILE_N * NVERT];      // 8 KB
    __shared__ float s_W   [NGAUSS * NGAUSS];     // 25.6 KB
    __shared__ float s_desc[KCHUNK * TILE_N * NGAUSS]; // 10 KB
    __shared__ float s_mr[NGAUSS], s_sri[NGAUSS], s_mt[NGAUSS], s_sti[NGAUSS], s_b[NGAUSS];

    const int tid = threadIdx.x;
    const int f   = blockIdx.y;
    const int n0  = blockIdx.x * TILE_N;

    // ---- stage sample tile ----
    for (int i = tid; i < TILE_N * NVERT; i += 256) {
        const int s = i >> 7, v = i & (NVERT - 1);
        const int gi = (n0 + s) * NVERT + v;
        const float m = mask[gi];
        // mask==0 -> poison rho so exp(-dr^2/s^2) underflows to exactly 0
        s_rho[i] = (m != 0.0f) ? rho[gi] : 1e19f;
        s_th [i] = theta[gi];
        s_ft [i] = feat[gi * NFEAT + f];
    }
    for (int i = tid; i < NGAUSS * NGAUSS; i += 256)
        s_W[i] = Wc[f * NGAUSS * NGAUSS + i];
    if (tid < NGAUSS) {
        s_mr[tid] = mu_rho[f * NGAUSS + tid];
        const float sr = sigma_rho[f * NGAUSS + tid];
        s_sri[tid] = 1.0f / (sr * sr + EPSF);
        s_mt[tid] = mu_theta[f * NGAUSS + tid];
        const float st = sigma_theta[f * NGAUSS + tid];
        s_sti[tid] = 1.0f / (st * st + EPSF);
        s_b[tid] = bc[f * NGAUSS + tid];
    }
    __syncthreads();

    const int lane = tid & 31;
    const int wave = tid >> 5;
    const int lrow = lane & 15;
    const int hi   = lane >> 4;

    v8f best;
    #pragma unroll
    for (int r = 0; r < 8; ++r) best[r] = -INFINITY;

    for (int kc = 0; kc < NROT; kc += KCHUNK) {
        // ---- accumulate num/den over verts; each thread owns 5 (s,g) pairs ----
        #pragma unroll
        for (int i = 0; i < 5; ++i) {
            const int p = tid + 256 * i;         // 0..1279
            const int s = p & 15;
            const int g = p >> 4;                // 0..79
            const float mr = s_mr[g], sri = s_sri[g];
            const float mt = s_mt[g], sti = s_sti[g];
            float nacc[KCHUNK], dacc[KCHUNK];
            #pragma unroll
            for (int kl = 0; kl < KCHUNK; ++kl) { nacc[kl] = 0.0f; dacc[kl] = 0.0f; }

            for (int v = 0; v < NVERT; ++v) {
                const float rv = s_rho[s * NVERT + v];
                const float dr = rv - mr;
                const float wb = __expf(-dr * dr * sri);   // rho gaussian * mask
                const float wf = wb * s_ft[s * NVERT + v];
                const float t0 = s_th[s * NVERT + v];
                #pragma unroll
                for (int kl = 0; kl < KCHUNK; ++kl) {
                    float th = t0 + (float)(kc + kl) * DTHETA;
                    th = (th >= TWO_PI_F) ? (th - TWO_PI_F) : th;   // mod 2pi
                    const float dt = th - mt;
                    const float tg = __expf(-dt * dt * sti);
                    nacc[kl] += wf * tg;
                    dacc[kl] += wb * tg;
                }
            }
            #pragma unroll
            for (int kl = 0; kl < KCHUNK; ++kl)
                s_desc[(kl * TILE_N + s) * NGAUSS + g] = nacc[kl] / (dacc[kl] + EPSF);
        }
        __syncthreads();

        // ---- conv GEMM: C(16x16) = desc(16x80) @ Wf(80x16-tile), fp32 WMMA ----
        if (wave < 5) {
            const int cb = wave * 16;
            #pragma unroll
            for (int kl = 0; kl < KCHUNK; ++kl) {
                v8f c;
                #pragma unroll
                for (int r = 0; r < 8; ++r) c[r] = 0.0f;
                #pragma unroll 4
                for (int kk = 0; kk < NGAUSS / 4; ++kk) {
                    const int kb = kk * 4 + 2 * hi;  // lanes 16-31 handle K+2,K+3
                    v2f a, b;
                    a.x = s_desc[(kl * TILE_N + lrow) * NGAUSS + kb];
                    a.y = s_desc[(kl * TILE_N + lrow) * NGAUSS + kb + 1];
                    b.x = s_W[(kb    ) * NGAUSS + cb + lrow];
                    b.y = s_W[(kb + 1) * NGAUSS + cb + lrow];
                    c = __builtin_amdgcn_wmma_f32_16x16x4_f32(
                            false, a, false, b, (short)0, c, false, false);
                }
                #pragma unroll
                for (int r = 0; r < 8; ++r) best[r] = fmaxf(best[r], c[r]);
            }
        }
        __syncthreads();
    }

    // ---- bias + relu + store to x (layout: n x (f*80+g')) ----
    if (wave < 5) {
        const int cb = wave * 16;
        #pragma unroll
        for (int r = 0; r < 8; ++r) {
            const int s  = r + 8 * hi;         // C/D layout: VGPR r = rows r, r+8
            const int gp = cb + lrow;
            const float v = fmaxf(best[r] + s_b[gp], 0.0f);
            x_out[(size_t)(n0 + s) * DIN + f * NGAUSS + gp] = v;
        }
    }
}

// -------------------------------------------------------------------------
// Kernel 2: MLP head  x(16x400)@W1 -> relu -> W2 -> relu -> W3 -> relu -> W4
// grid = 625, block = 32 threads (one wave32)
// -------------------------------------------------------------------------
__global__ __launch_bounds__(32) void masif_mlp_kernel(
    const float* __restrict__ x,
    const float* __restrict__ W1, const float* __restrict__ b1,
    const float* __restrict__ W2, const float* __restrict__ b2,
    const float* __restrict__ W3, const float* __restrict__ b3,
    const float* __restrict__ W4, const float* __restrict__ b4,
    float* __restrict__ out)
{
    __shared__ float s_h1[TILE_N][NGAUSS];

    const int lane = threadIdx.x & 31;
    const int lrow = lane & 15;
    const int hi   = lane >> 4;
    const int n0   = blockIdx.x * TILE_N;

    for (int t = 0; t < 5; ++t) {          // 5 column tiles of 16 -> 80 outputs
        v8f c;
        #pragma unroll
        for (int r = 0; r < 8; ++r) c[r] = 0.0f;
        #pragma unroll 4
        for (int kk = 0; kk < DIN / 4; ++kk) {   // 100 K-steps of K=4
            const int kb = kk * 4 + 2 * hi;
            v2f a, b;
            const size_t arow = (size_t)(n0 + lrow) * DIN + kb;
            a.x = x[arow];
            a.y = x[arow + 1];
            b.x = W1[(kb    ) * NGAUSS + t * 16 + lrow];
            b.y = W1[(kb + 1) * NGAUSS + t * 16 + lrow];
            c = __builtin_amdgcn_wmma_f32_16x16x4_f32(
                    false, a, false, b, (short)0, c, false, false);
        }
        #pragma unroll
        for (int r = 0; r < 8; ++r) {
            const int s   = r + 8 * hi;
            const int col = t * 16 + lrow;
            s_h1[s][col] = fmaxf(c[r] + b1[col], 0.0f);
        }
    }
    __syncthreads();

    if (lane < 16) {
        const int s = lane;
        float h2[5];
        #pragma unroll
        for (int j = 0; j < 5; ++j) {
            float acc = b2[j];
            for (int g = 0; g < NGAUSS; ++g) acc += s_h1[s][g] * W2[g * 5 + j];
            h2[j] = fmaxf(acc, 0.0f);
        }
        float h3[16];
        #pragma unroll
        for (int m = 0; m < 16; ++m) {
            float acc = b3[m];
            #pragma unroll
            for (int j = 0; j < 5; ++j) acc += h2[j] * W3[j * 16 + m];
            h3[m] = fmaxf(acc, 0.0f);
        }
        #pragma unroll
        for (int c2 = 0; c2 < 2; ++c2) {
            float acc = b4[c2];
            #pragma unroll
            for (int m = 0; m < 16; ++m) acc += h3[m] * W4[m * 2 + c2];
            out[(size_t)(n0 + s) * 2 + c2] = acc;
        }
    }
}

// -------------------------------------------------------------------------
extern "C" void kernel_launch(void* const* d_in, const int* in_sizes, int n_in,
                              void* d_out, int out_size, void* d_ws, size_t ws_size,
                              hipStream_t stream) {
    (void)in_sizes; (void)n_in; (void)out_size; (void)ws_size;
    const float* rho_c   = (const float*)d_in[0];
    const float* theta_c = (const float*)d_in[1];
    const float* in_feat = (const float*)d_in[2];
    const float* mask    = (const float*)d_in[3];
    const float* mu_rho  = (const float*)d_in[4];
    const float* sig_rho = (const float*)d_in[5];
    const float* mu_th   = (const float*)d_in[6];
    const float* sig_th  = (const float*)d_in[7];
    const float* W_conv  = (const float*)d_in[8];
    const float* b_conv  = (const float*)d_in[9];
    const float* W1      = (const float*)d_in[10];
    const float* b1      = (const float*)d_in[11];
    const float* W2      = (const float*)d_in[12];
    const float* b2      = (const float*)d_in[13];
    const float* W3      = (const float*)d_in[14];
    const float* b3      = (const float*)d_in[15];
    const float* W4      = (const float*)d_in[16];
    const float* b4      = (const float*)d_in[17];

    float* x = (float*)d_ws;                      // 10000 x 400 fp32 = 16 MB

    dim3 g1(NSAMP / TILE_N, NFEAT);               // 625 x 5
    masif_gauss_conv_kernel<<<g1, 256, 0, stream>>>(
        rho_c, theta_c, in_feat, mask, mu_rho, sig_rho, mu_th, sig_th,
        W_conv, b_conv, x);

    masif_mlp_kernel<<<dim3(NSAMP / TILE_N), 32, 0, stream>>>(
        x, W1, b1, W2, b2, W3, b3, W4, b4, (float*)d_out);
}